// RNN_11862699671577
// MI455X (gfx1250) — compile-verified
//
#include <hip/hip_runtime.h>
#include <hip/hip_bf16.h>
#include <math.h>

// CDNA5 / gfx1250 LSTM:
//  Phase 1: x_gates = x @ W_ih^T + (b_ih+b_hh) via V_WMMA_F32_16X16X4_F32
//           (A = 16 timesteps x K=4 of x, B = 4x16 = W_ih^T, C = bias bcast).
//  Phase 2: single-wave sequential scan; W_hh + h/c state live in registers,
//           cross-lane traffic via ds_bpermute (__shfl), x_gates streamed with
//           coalesced 128B wave loads + global_prefetch_b8 lookahead.

typedef __attribute__((ext_vector_type(2))) float v2f;
typedef __attribute__((ext_vector_type(8))) float v8f;

#define HID 4
#define NG  16  // 4*H gates

// ---------------------------------------------------------------------------
// Phase 1: one wave32 -> one 16-timestep tile via one WMMA.
//   A(16x4) f32 layout: lanes 0-15 hold M=lane, VGPR0/1 = K0/K1;
//                       lanes 16-31 hold M=lane-16, VGPR0/1 = K2/K3.
//   B(4x16) mirrored:   lane half 0 -> rows K0/K1 col n; half 1 -> K2/K3.
//   C/D(16x16): VGPR r: lanes 0-15 = (M=r, N=lane), lanes 16-31 = (M=8+r).
// ---------------------------------------------------------------------------
__global__ __launch_bounds__(256) void lstm_gates_wmma(
    const float* __restrict__ x,      // [T,4]
    const float* __restrict__ W_ih,   // [16,4]
    const float* __restrict__ b_ih,   // [16]
    const float* __restrict__ b_hh,   // [16]
    float* __restrict__ xg)           // [T,16] out
{
  const int lane = threadIdx.x & 31;
  const int wave = blockIdx.x * (blockDim.x >> 5) + (threadIdx.x >> 5);
  const size_t t0 = (size_t)wave * 16;

  const int n    = lane & 15;   // tile row (A) / column (B,C,D)
  const int half = lane >> 4;   // K-pair selector

  // A: x rows t0..t0+15, K-pair per half-wave (64-bit coalesced load)
  const float* ap = x + (t0 + (size_t)n) * HID + half * 2;
  v2f a; a.x = ap[0]; a.y = ap[1];

  // B[k][c] = W_ih[c][k]  (W_ih is [16,4] row-major -> rows are K-vectors)
  const float* bp = W_ih + n * HID + half * 2;
  v2f b; b.x = bp[0]; b.y = bp[1];

  // C: bias broadcast down each column
  const float bias = b_ih[n] + b_hh[n];
  v8f c;
#pragma unroll
  for (int r = 0; r < 8; ++r) c[r] = bias;

  // D = A*B + C   (8 args: neg_a, A, neg_b, B, c_mod, C, reuse_a, reuse_b)
  v8f d = __builtin_amdgcn_wmma_f32_16x16x4_f32(
      false, a, false, b, (short)0, c, false, false);

  // Scatter D: vgpr r -> xg[t0 + half*8 + r][n]
  float* op = xg + (t0 + (size_t)half * 8) * NG + n;
#pragma unroll
  for (int r = 0; r < 8; ++r) op[(size_t)r * NG] = d[r];
}

// ---------------------------------------------------------------------------
// Phase 2: single wave32 sequential scan over T steps.
//   lane g (0..15, mirrored on 16..31) owns gate row g of W_hh.
//   lane column j = g&3 redundantly tracks c[j]; h broadcast via shfl.
// ---------------------------------------------------------------------------
__global__ __launch_bounds__(32) void lstm_scan(
    const float* __restrict__ xg,     // [T,16]
    const float* __restrict__ W_hh,   // [16,4]
    const float* __restrict__ h0,     // [4]
    const float* __restrict__ c0,     // [4]
    float* __restrict__ out,          // [T,4]
    int T)
{
  const int lane = threadIdx.x;       // 0..31
  const int g    = lane & 15;         // gate row this lane owns
  const int j    = g & 3;             // hidden column this lane tracks
  const int quad = g >> 2;            // 0:i 1:f 2:g 3:o

  // W_hh row for this gate (stays in VGPRs for the whole scan)
  const float w0 = W_hh[g * HID + 0];
  const float w1 = W_hh[g * HID + 1];
  const float w2 = W_hh[g * HID + 2];
  const float w3 = W_hh[g * HID + 3];

  float h0v = h0[0], h1v = h0[1], h2v = h0[2], h3v = h0[3];
  float cst = c0[j];                  // c state for column j (replicated)

  const size_t total = (size_t)T * NG;

  for (int t = 0; t < T; t += 2) {
    // One coalesced 128B wave load covers timesteps t and t+1.
    const size_t base = (size_t)t * NG;
    const float xv = xg[base + lane];

    // Run ~4KB ahead of the serial chain (global_prefetch_b8; speculative,
    // OOB prefetches are silently dropped).
    {
      size_t pf = base + 1024 + lane;
      if (pf < total) __builtin_prefetch(xg + pf, 0, 1);
    }

#pragma unroll
    for (int s = 0; s < 2; ++s) {
      // Pull this step's pre-activation for gate g out of the wave load.
      const float xval = __shfl(xv, s * NG + g, 32);

      // gate = xg + W_hh[g,:] . h   (4-FMA chain, hidden GEMV)
      const float gate = xval + w0 * h0v + w1 * h1v + w2 * h2v + w3 * h3v;

      // i,f,o -> sigmoid; g -> tanh (HW trans units: v_exp/v_tanh)
      const float act = (quad == 2) ? tanhf(gate)
                                    : 1.0f / (1.0f + __expf(-gate));

      // Gather (i,f,g,o) for column j across lanes.
      const float iv = __shfl(act, j + 0,  32);
      const float fv = __shfl(act, j + 4,  32);
      const float gv = __shfl(act, j + 8,  32);
      const float ov = __shfl(act, j + 12, 32);

      cst = fv * cst + iv * gv;
      const float hj = ov * tanhf(cst);

      // Broadcast h_new to every lane for the next GEMV.
      h0v = __shfl(hj, 0, 32);
      h1v = __shfl(hj, 1, 32);
      h2v = __shfl(hj, 2, 32);
      h3v = __shfl(hj, 3, 32);

      if (lane < 4) out[(size_t)(t + s) * HID + j] = hj;
    }
  }
}

// ---------------------------------------------------------------------------
extern "C" void kernel_launch(void* const* d_in, const int* in_sizes, int n_in,
                              void* d_out, int out_size, void* d_ws, size_t ws_size,
                              hipStream_t stream) {
  const float* x    = (const float*)d_in[0];  // [T,4]
  const float* W_ih = (const float*)d_in[1];  // [16,4]
  const float* W_hh = (const float*)d_in[2];  // [16,4]
  const float* b_ih = (const float*)d_in[3];  // [16]
  const float* b_hh = (const float*)d_in[4];  // [16]
  const float* h0   = (const float*)d_in[5];  // [4]
  const float* c0   = (const float*)d_in[6];  // [4]
  float* out = (float*)d_out;

  const int T = in_sizes[0] / HID;            // 1,048,576
  float* xg = (float*)d_ws;                   // [T,16] scratch (64 MB)

  // Phase 1: 1 wave per 16 timesteps; 8 waves (256 thr) per block.
  const int tiles  = T / 16;
  const int blocks = tiles / 8;
  lstm_gates_wmma<<<blocks, 256, 0, stream>>>(x, W_ih, b_ih, b_hh, xg);

  // Phase 2: single-wave sequential scan (stream-ordered after phase 1).
  lstm_scan<<<1, 32, 0, stream>>>(xg, W_hh, h0, c0, out, T);
}